// LinearGaussianQ_53987738911227
// MI455X (gfx1250) — compile-verified
//
#include <hip/hip_runtime.h>
#include <math.h>

#define LOG2PI_F 1.8378770664093453f

typedef float v2f __attribute__((ext_vector_type(2)));
typedef float v8f __attribute__((ext_vector_type(8)));

// ---- workspace layout (float offsets) ----
#define OFF_B     0         // 256*256   B_t
#define OFF_A     65536     // 256*16    a_t
#define OFF_BCOV  69632     // 256*256   bcov_t
#define OFF_OM    135168    // 3*256     Om0, Om_obs, Om_tr
#define OFF_MF    135936    // 16        final m_f
#define OFF_PF    135952    // 256       final P_f
#define OFF_AS    136208    // 512*256   final As
#define OFF_BS    267280    // 512*16    final bs

__device__ __forceinline__ float cterm16(float logdet) {
  return -0.5f * (16.f * LOG2PI_F + logdet);
}

// Gauss-Jordan inverse of 16x16 (no pivoting; inputs are SPD-ish).
// M destroyed, Inv = M^-1, *ld += log|det|. 256 threads, tid = r*16+c.
__device__ void gj_invert16(float* M, float* Inv, float* ld, int tid) {
  const int r = tid >> 4, c = tid & 15;
  Inv[tid] = (r == c) ? 1.f : 0.f;
  __syncthreads();
  for (int k = 0; k < 16; ++k) {
    float piv = M[k * 16 + k];
    if (tid == 0) *ld += logf(fabsf(piv));
    float ip = 1.f / piv;
    __syncthreads();
    if (r == k) { M[tid] *= ip; Inv[tid] *= ip; }
    __syncthreads();
    float f  = M[r * 16 + k];
    float mk = M[k * 16 + c];
    float ik = Inv[k * 16 + c];
    __syncthreads();
    if (r != k) { M[tid] -= f * mk; Inv[tid] -= f * ik; }
    __syncthreads();
  }
}

// Kalman measurement update under q. sm/sP updated in place.
__device__ void kupdate16(int tid, const float* y,
    const float* q_em_w, const float* q_em_b, const float* q_em_cov,
    float* sm, float* sP, float* sM, float* sInv,
    float* sT1, float* sT2, float* sT3, float* sv, float* ld)
{
  const int r = tid >> 4, c = tid & 15;
  float s;
  // T1 = q_em_w @ P
  s = 0.f; for (int l = 0; l < 16; ++l) s += q_em_w[r * 16 + l] * sP[l * 16 + c];
  __syncthreads(); sT1[tid] = s; __syncthreads();
  // S = T1 @ q_em_w^T + q_em_cov
  s = q_em_cov[tid]; for (int l = 0; l < 16; ++l) s += sT1[r * 16 + l] * q_em_w[c * 16 + l];
  __syncthreads(); sM[tid] = s; if (tid == 0) *ld = 0.f; __syncthreads();
  gj_invert16(sM, sInv, ld, tid);                     // sInv = S^-1
  // T2 = P @ q_em_w^T
  s = 0.f; for (int l = 0; l < 16; ++l) s += sP[r * 16 + l] * q_em_w[c * 16 + l];
  __syncthreads(); sT2[tid] = s; __syncthreads();
  // Kg = T2 @ Sinv  -> sT3
  s = 0.f; for (int l = 0; l < 16; ++l) s += sT2[r * 16 + l] * sInv[l * 16 + c];
  __syncthreads(); sT3[tid] = s; __syncthreads();
  // innovation
  float vv = 0.f;
  if (tid < 16) { vv = y[tid] - q_em_b[tid]; for (int j = 0; j < 16; ++j) vv -= q_em_w[tid * 16 + j] * sm[j]; }
  __syncthreads(); if (tid < 16) sv[tid] = vv; __syncthreads();
  // m += Kg @ innovation
  float mm2 = 0.f;
  if (tid < 16) { mm2 = sm[tid]; for (int j = 0; j < 16; ++j) mm2 += sT3[tid * 16 + j] * sv[j]; }
  __syncthreads(); if (tid < 16) sm[tid] = mm2; __syncthreads();
  // T1 = Kg @ q_em_w
  s = 0.f; for (int l = 0; l < 16; ++l) s += sT3[r * 16 + l] * q_em_w[l * 16 + c];
  __syncthreads(); sT1[tid] = s; __syncthreads();
  // T2 = I - T1
  sT2[tid] = ((r == c) ? 1.f : 0.f) - sT1[tid]; __syncthreads();
  // P = T2 @ P
  s = 0.f; for (int l = 0; l < 16; ++l) s += sT2[r * 16 + l] * sP[l * 16 + c];
  __syncthreads(); sM[tid] = s; __syncthreads();
  sP[tid] = sM[tid]; __syncthreads();
}

// ---------------- K1: sequential precompute (1 block, 256 threads) -------
__global__ __launch_bounds__(256) void lgq_k1(
    const float* obs,
    const float* p_prior_mean, const float* p_prior_cov,
    const float* p_trans_w, const float* p_trans_b, const float* p_trans_cov,
    const float* p_em_w, const float* p_em_b, const float* p_em_cov,
    const float* q_prior_mean, const float* q_prior_cov,
    const float* q_trans_w, const float* q_trans_b, const float* q_trans_cov,
    const float* q_em_w, const float* q_em_b, const float* q_em_cov,
    float* ws, float* out)
{
  __shared__ float sM[256], sInv[256], sT1[256], sT2[256], sT3[256];
  __shared__ float sQtp[256], sCm[256], sCtC[256], sP[256], sPprec[256];
  __shared__ float sm[16], scq[16], sv[16];
  __shared__ float sld, sconst;
  const int tid = threadIdx.x;
  const int r = tid >> 4, c = tid & 15;
  float s;
  (void)p_em_w; (void)p_em_b; (void)p_trans_b; (void)p_prior_mean;

  if (tid == 0) { sconst = 0.f; sld = 0.f; }
  // Om_tr = -0.5 * inv(p_trans_cov)
  sM[tid] = p_trans_cov[tid]; __syncthreads();
  gj_invert16(sM, sInv, &sld, tid);
  ws[OFF_OM + 512 + tid] = -0.5f * sInv[tid];
  float ld_ptr = sld; __syncthreads();
  if (tid == 0) sld = 0.f;
  // Om_obs = -0.5 * inv(p_em_cov)
  sM[tid] = p_em_cov[tid]; __syncthreads();
  gj_invert16(sM, sInv, &sld, tid);
  ws[OFF_OM + 256 + tid] = -0.5f * sInv[tid];
  float ld_pem = sld; __syncthreads();
  if (tid == 0) sld = 0.f;
  // Om0 = -0.5 * inv(p_prior_cov)
  sM[tid] = p_prior_cov[tid]; __syncthreads();
  gj_invert16(sM, sInv, &sld, tid);
  ws[OFF_OM + tid] = -0.5f * sInv[tid];
  float ld_pprior = sld; __syncthreads();
  if (tid == 0) sld = 0.f;
  // q_tr_prec
  sM[tid] = q_trans_cov[tid]; __syncthreads();
  gj_invert16(sM, sInv, &sld, tid);
  sQtp[tid] = sInv[tid]; __syncthreads();
  // cm = q_trans_w^T @ q_tr_prec
  s = 0.f; for (int l = 0; l < 16; ++l) s += q_trans_w[l * 16 + r] * sQtp[l * 16 + c];
  __syncthreads(); sCm[tid] = s; __syncthreads();
  // CtC = cm @ q_trans_w
  s = 0.f; for (int l = 0; l < 16; ++l) s += sCm[r * 16 + l] * q_trans_w[l * 16 + c];
  __syncthreads(); sCtC[tid] = s; __syncthreads();
  // cq = cm @ q_trans_b
  float cqv = 0.f;
  if (tid < 16) { for (int j = 0; j < 16; ++j) cqv += sCm[tid * 16 + j] * q_trans_b[j]; }
  __syncthreads(); if (tid < 16) scq[tid] = cqv; __syncthreads();
  // Kalman init under q
  sP[tid] = q_prior_cov[tid];
  if (tid < 16) sm[tid] = q_prior_mean[tid];
  __syncthreads();
  kupdate16(tid, obs, q_em_w, q_em_b, q_em_cov, sm, sP, sM, sInv, sT1, sT2, sT3, sv, &sld);
  if (tid == 0) sconst = cterm16(ld_pprior) + cterm16(ld_pem);
  __syncthreads();

  for (int t = 1; t < 256; ++t) {
    // P_prec = inv(P_f)
    sM[tid] = sP[tid]; if (tid == 0) sld = 0.f; __syncthreads();
    gj_invert16(sM, sInv, &sld, tid);
    sPprec[tid] = sInv[tid]; __syncthreads();
    // bcov = inv(CtC + P_prec)
    sM[tid] = sCtC[tid] + sPprec[tid]; if (tid == 0) sld = 0.f; __syncthreads();
    gj_invert16(sM, sInv, &sld, tid);
    float ld_bprec = sld;                  // log det(bprec) = -log det(bcov)
    ws[OFF_BCOV + t * 256 + tid] = sInv[tid];
    __syncthreads();
    // B = bcov @ cm
    s = 0.f; for (int l = 0; l < 16; ++l) s += sInv[r * 16 + l] * sCm[l * 16 + c];
    ws[OFF_B + t * 256 + tid] = s;
    // a = bcov @ (P_prec @ m_f - cq)
    float w1 = 0.f;
    if (tid < 16) { w1 = -scq[tid]; for (int j = 0; j < 16; ++j) w1 += sPprec[tid * 16 + j] * sm[j]; }
    __syncthreads(); if (tid < 16) sv[tid] = w1; __syncthreads();
    float av = 0.f;
    if (tid < 16) { for (int j = 0; j < 16; ++j) av += sInv[tid * 16 + j] * sv[j];
                    ws[OFF_A + t * 16 + tid] = av; }
    if (tid == 0)
      sconst += cterm16(ld_pem) + cterm16(ld_ptr) - cterm16(-ld_bprec) + 8.f;
    __syncthreads();
    // Kalman predict
    float mp = 0.f;
    if (tid < 16) { mp = q_trans_b[tid]; for (int j = 0; j < 16; ++j) mp += q_trans_w[tid * 16 + j] * sm[j]; }
    __syncthreads(); if (tid < 16) sm[tid] = mp; __syncthreads();
    s = 0.f; for (int l = 0; l < 16; ++l) s += q_trans_w[r * 16 + l] * sP[l * 16 + c];
    __syncthreads(); sT1[tid] = s; __syncthreads();
    s = q_trans_cov[tid]; for (int l = 0; l < 16; ++l) s += sT1[r * 16 + l] * q_trans_w[c * 16 + l];
    __syncthreads(); sP[tid] = s; __syncthreads();
    kupdate16(tid, obs + t * 16, q_em_w, q_em_b, q_em_cov, sm, sP, sM, sInv, sT1, sT2, sT3, sv, &sld);
  }
  if (tid < 16) ws[OFF_MF + tid] = sm[tid];
  ws[OFF_PF + tid] = sP[tid];
  sM[tid] = sP[tid]; if (tid == 0) sld = 0.f; __syncthreads();
  gj_invert16(sM, sInv, &sld, tid);
  if (tid == 0) out[0] = sconst - cterm16(sld) + 8.f;   // - cterm(dz,det P_f) + 0.5*dz
}

// ---- one integration step for a slot: 16 chained f32 WMMAs --------------
// FIX: subtract I from AB after the multiply (creation step of even slots)
template <bool FIX>
__device__ __forceinline__ void lgq_step(
    float* A, const float* Bt, const float* Ct, const float* at,
    const v2f* OmOp, v8f& bsacc, float& acc,
    int n, int half, int kk, int lane)
{
  v2f Aop[4], Cb[4], Bb[4], Ab[4], ab[4];
#pragma unroll
  for (int c0 = 0; c0 < 4; ++c0) {
    // A as A-operand (rows in lanes, K split over halves)
    Aop[c0].x = A[n * 16 + 4 * c0 + kk];
    Aop[c0].y = A[n * 16 + 4 * c0 + kk + 1];
    // B-operands (rows striped across lanes)
    int r0 = 4 * c0 + 2 * half;
    Cb[c0].x = Ct[r0 * 16 + n];       Cb[c0].y = Ct[(r0 + 1) * 16 + n];
    Bb[c0].x = Bt[r0 * 16 + n];       Bb[c0].y = Bt[(r0 + 1) * 16 + n];
    Ab[c0].x = A[r0 * 16 + n];        Ab[c0].y = A[(r0 + 1) * 16 + n];
    // vector a as column-0 matrix: branchless (load + cndmask)
    float ax = at[r0], ay = at[r0 + 1];
    ab[c0].x = (n == 0) ? ax : 0.f;
    ab[c0].y = (n == 0) ? ay : 0.f;
  }
  const v8f vzero = {0.f, 0.f, 0.f, 0.f, 0.f, 0.f, 0.f, 0.f};
  v8f AC = vzero, OA = vzero, AB = vzero;
#pragma unroll
  for (int c0 = 0; c0 < 4; ++c0) {
    AC    = __builtin_amdgcn_wmma_f32_16x16x4_f32(false, Aop[c0],  false, Cb[c0], (short)0, AC,    false, false);
    OA    = __builtin_amdgcn_wmma_f32_16x16x4_f32(false, OmOp[c0], false, Ab[c0], (short)0, OA,    false, false);
    AB    = __builtin_amdgcn_wmma_f32_16x16x4_f32(false, Aop[c0],  false, Bb[c0], (short)0, AB,    false, false);
    bsacc = __builtin_amdgcn_wmma_f32_16x16x4_f32(false, Aop[c0],  false, ab[c0], (short)0, bsacc, false, false);
  }
  // const contribution: sum (A@bcov) .* (Om@A)  == trace(Om A bcov A^T)
#pragma unroll
  for (int i = 0; i < 8; ++i) acc += AC[i] * OA[i];
  if (FIX) {
#pragma unroll
    for (int i = 0; i < 8; ++i)
      if (lane == i || lane == 24 + i) AB[i] -= 1.f;
  }
  // write back A <- A @ B (D layout -> row-major LDS)
#pragma unroll
  for (int i = 0; i < 8; ++i) A[(half * 8 + i) * 16 + n] = AB[i];
}

// ---------------- K2: per-slot WMMA recursion (wave per slot) ------------
__global__ __launch_bounds__(256) void lgq_k2(
    const float* obs, const float* p_prior_mean,
    const float* p_trans_w, const float* p_trans_b,
    const float* p_em_w, const float* p_em_b,
    const float* ws, float* As_g, float* bs_g, float* out)
{
  __shared__ float stage[2048];                 // 8 waves x 16x16 staging
  const int lane = threadIdx.x & 31;
  const int wv   = threadIdx.x >> 5;
  const int k    = blockIdx.x * 8 + wv;         // slot 0..511
  float* A = stage + wv * 256;

  const float* wsB  = ws + OFF_B;
  const float* wsa  = ws + OFF_A;
  const float* wsC  = ws + OFF_BCOV;
  const float* wsOm = ws + OFF_OM;

  const int t0 = (k == 0) ? 1 : ((k + 1) >> 1);
  const bool dead = (t0 > 255);                 // slot 511 is never populated

  // initial A (row-major in LDS): slot0=I, odd=p_em_w, even=p_trans_w
  {
    const float* src = (k & 1) ? p_em_w : p_trans_w;
    for (int i = 0; i < 8; ++i) {
      int e = lane + 32 * i;
      float v;
      if (dead) v = 0.f;
      else if (k == 0) v = ((e >> 4) == (e & 15)) ? 1.f : 0.f;
      else v = src[e];
      A[e] = v;
    }
  }
  const int n    = lane & 15;
  const int half = lane >> 4;
  const int kk   = half * 2;

  // Om in A-operand layout (Om symmetric)
  const float* Om = wsOm + ((k == 0) ? 0 : ((k & 1) ? 256 : 512));
  v2f OmOp[4];
#pragma unroll
  for (int c0 = 0; c0 < 4; ++c0) {
    OmOp[c0].x = Om[n * 16 + 4 * c0 + kk];
    OmOp[c0].y = Om[n * 16 + 4 * c0 + kk + 1];
  }
  const v8f vzero = {0.f, 0.f, 0.f, 0.f, 0.f, 0.f, 0.f, 0.f};
  v8f bsacc = vzero;
  float acc = 0.f;

  if (!dead) {
    // creation step (peeled): even slots k>=2 need the -I fixup
    if (k != 0 && (k & 1) == 0)
      lgq_step<true >(A, wsB + t0 * 256, wsC + t0 * 256, wsa + t0 * 16,
                      OmOp, bsacc, acc, n, half, kk, lane);
    else
      lgq_step<false>(A, wsB + t0 * 256, wsC + t0 * 256, wsa + t0 * 16,
                      OmOp, bsacc, acc, n, half, kk, lane);
    // steady state: branch-free body + prefetch of next step's tiles
    for (int t = t0 + 1; t < 256; ++t) {
      __builtin_prefetch(wsB + (t + 1) * 256 + lane * 8, 0, 0);
      __builtin_prefetch(wsC + (t + 1) * 256 + lane * 8, 0, 0);
      __builtin_prefetch(wsa + (t + 1) * 16 + n, 0, 0);
      lgq_step<false>(A, wsB + t * 256, wsC + t * 256, wsa + t * 16,
                      OmOp, bsacc, acc, n, half, kk, lane);
    }
  }
  // final As to global
#pragma unroll
  for (int i = 0; i < 8; ++i) {
    int row = half * 8 + i;
    As_g[k * 256 + row * 16 + n] = A[row * 16 + n];
  }
  // bs = b_init + column 0 of bsacc (lanes 0 and 16)
  if (n == 0) {
#pragma unroll
    for (int i = 0; i < 8; ++i) {
      int mrow = half * 8 + i;
      float bi;
      if (dead) bi = 0.f;
      else if (k == 0) bi = -p_prior_mean[mrow];
      else if (k & 1) bi = p_em_b[mrow] - obs[(t0 - 1) * 16 + mrow];
      else bi = p_trans_b[mrow];
      bs_g[k * 16 + mrow] = bi + bsacc[i];
    }
  }
  for (int off = 16; off; off >>= 1) acc += __shfl_xor(acc, off, 32);
  if (lane == 0) atomicAdd(out, acc);
}

// ---------------- K3: final expectation under filtering ------------------
__global__ __launch_bounds__(256) void lgq_k3(
    const float* obs, const float* p_em_w, const float* p_em_b,
    const float* ws, const float* As_g, const float* bs_g, float* out)
{
  __shared__ float sA[256], sOm[256], sP[256], sM1[256], sM2[256];
  __shared__ float smv[16], su[16], red[8];
  const int tid = threadIdx.x;
  const int r = tid >> 4, c = tid & 15;
  const int k = blockIdx.x;                    // 0..512 (512 = pending obs term)
  const float* wsOm = ws + OFF_OM;

  sP[tid] = ws[OFF_PF + tid];
  if (tid < 16) smv[tid] = ws[OFF_MF + tid];
  sA[tid] = (k < 512) ? As_g[k * 256 + tid] : p_em_w[tid];
  int par = (k == 0) ? 0 : ((k & 1) ? 1 : 2);
  if (k == 512) par = 1;                       // Om_obs
  sOm[tid] = wsOm[par * 256 + tid];
  __syncthreads();
  // M1 = A @ P_f
  float s = 0.f; for (int l = 0; l < 16; ++l) s += sA[r * 16 + l] * sP[l * 16 + c];
  __syncthreads(); sM1[tid] = s; __syncthreads();
  // M2 = M1 @ A   (no transpose: faithful to reference)
  s = 0.f; for (int l = 0; l < 16; ++l) s += sM1[r * 16 + l] * sA[l * 16 + c];
  sM2[tid] = s;
  // u = A @ m_f + b
  float uu = 0.f;
  if (tid < 16) {
    float bb = (k < 512) ? bs_g[k * 16 + tid] : (p_em_b[tid] - obs[255 * 16 + tid]);
    uu = bb;
    for (int j = 0; j < 16; ++j) uu += sA[tid * 16 + j] * smv[j];
  }
  __syncthreads(); if (tid < 16) su[tid] = uu; __syncthreads();
  // contribution = sum_{ij} Om_ij * ( (A P A)_{ji} + u_i u_j )
  float contrib = sOm[r * 16 + c] * (sM2[c * 16 + r] + su[r] * su[c]);
  for (int off = 16; off; off >>= 1) contrib += __shfl_xor(contrib, off, 32);
  if ((tid & 31) == 0) red[tid >> 5] = contrib;
  __syncthreads();
  if (tid < 8) {
    float v = red[tid];
    for (int off = 4; off; off >>= 1) v += __shfl_xor(v, off, 32);
    if (tid == 0) atomicAdd(out, v);
  }
}

extern "C" void kernel_launch(void* const* d_in, const int* in_sizes, int n_in,
                              void* d_out, int out_size, void* d_ws, size_t ws_size,
                              hipStream_t stream) {
  (void)in_sizes; (void)n_in; (void)out_size; (void)ws_size;
  const float* obs          = (const float*)d_in[0];
  const float* p_prior_mean = (const float*)d_in[1];
  const float* p_prior_cov  = (const float*)d_in[2];
  const float* p_trans_w    = (const float*)d_in[3];
  const float* p_trans_b    = (const float*)d_in[4];
  const float* p_trans_cov  = (const float*)d_in[5];
  const float* p_em_w       = (const float*)d_in[6];
  const float* p_em_b       = (const float*)d_in[7];
  const float* p_em_cov     = (const float*)d_in[8];
  const float* q_prior_mean = (const float*)d_in[9];
  const float* q_prior_cov  = (const float*)d_in[10];
  const float* q_trans_w    = (const float*)d_in[11];
  const float* q_trans_b    = (const float*)d_in[12];
  const float* q_trans_cov  = (const float*)d_in[13];
  const float* q_em_w       = (const float*)d_in[14];
  const float* q_em_b       = (const float*)d_in[15];
  const float* q_em_cov     = (const float*)d_in[16];
  float* ws   = (float*)d_ws;
  float* out  = (float*)d_out;
  float* As_g = ws + OFF_AS;
  float* bs_g = ws + OFF_BS;

  lgq_k1<<<1, 256, 0, stream>>>(obs, p_prior_mean, p_prior_cov, p_trans_w, p_trans_b,
      p_trans_cov, p_em_w, p_em_b, p_em_cov, q_prior_mean, q_prior_cov, q_trans_w,
      q_trans_b, q_trans_cov, q_em_w, q_em_b, q_em_cov, ws, out);
  lgq_k2<<<64, 256, 0, stream>>>(obs, p_prior_mean, p_trans_w, p_trans_b, p_em_w, p_em_b,
      ws, As_g, bs_g, out);
  lgq_k3<<<513, 256, 0, stream>>>(obs, p_em_w, p_em_b, ws, As_g, bs_g, out);
}